// TransRModel_14800457302462
// MI455X (gfx1250) — compile-verified
//
#include <hip/hip_runtime.h>
#include <hip/hip_bf16.h>

#define E_DIM 128
#define R_DIM 64

typedef float v2f __attribute__((ext_vector_type(2)));
typedef float v8f __attribute__((ext_vector_type(8)));

// ---------------- counting-sort helpers ----------------

__global__ void k_zero(int* counts, int* cursors, int n) {
    int i = blockIdx.x * blockDim.x + threadIdx.x;
    if (i < n) { counts[i] = 0; cursors[i] = 0; }
}

__global__ void k_hist(const int* __restrict__ rel, int* __restrict__ counts, int B) {
    int i = blockIdx.x * blockDim.x + threadIdx.x;
    if (i < B) atomicAdd(&counts[rel[i]], 1);
}

__global__ void k_scan(const int* __restrict__ counts, int* __restrict__ offsets, int n) {
    if (blockIdx.x == 0 && threadIdx.x == 0) {
        int s = 0;
        for (int i = 0; i < n; ++i) { offsets[i] = s; s += counts[i]; }
        offsets[n] = s;
    }
}

__global__ void k_scatter(const int* __restrict__ rel, const int* __restrict__ offsets,
                          int* __restrict__ cursors, int* __restrict__ sorted, int B) {
    int i = blockIdx.x * blockDim.x + threadIdx.x;
    if (i < B) {
        int r = rel[i];
        int p = offsets[r] + atomicAdd(&cursors[r], 1);
        sorted[p] = i;
    }
}

// ---------------- main WMMA kernel: one workgroup per relation ----------------
// distance[b] = sum_c | ((h_b - t_b) . M_rel)[c] + r_rel[c] |

__global__ __launch_bounds__(256)
void transr_wmma(const int* __restrict__ head, const int* __restrict__ tail,
                 const float* __restrict__ entity_emb,
                 const float* __restrict__ relation_emb,
                 const float* __restrict__ proj_emb,
                 const int* __restrict__ offsets,
                 const int* __restrict__ sorted_idx,
                 float* __restrict__ out) {
    __shared__ float Ms[E_DIM * R_DIM];   // 32 KB: M[e][c] at e*64 + c
    __shared__ float Rs[R_DIM];

    const int rel   = blockIdx.x;
    const int start = offsets[rel];
    const int end   = offsets[rel + 1];
    const int count = end - start;
    if (count == 0) return;

    // Cooperative load of M tile into LDS (float4 vectorized: 2048 x 16B).
    {
        const float4* Mg = (const float4*)(proj_emb + (size_t)rel * (E_DIM * R_DIM));
        float4* Ml = (float4*)Ms;
        for (int i = threadIdx.x; i < (E_DIM * R_DIM) / 4; i += blockDim.x)
            Ml[i] = Mg[i];
        if (threadIdx.x < R_DIM)
            Rs[threadIdx.x] = relation_emb[(size_t)rel * R_DIM + threadIdx.x];
    }
    __syncthreads();

    const int lane  = threadIdx.x & 31;
    const int wave  = threadIdx.x >> 5;
    const int row   = lane & 15;   // A-matrix row / B,C column within tile
    const int khalf = lane >> 4;   // which K pair this lane holds (A layout)

    const int nchunks = (count + 15) >> 4;

    for (int c = wave; c < nchunks; c += 8) {          // wave-uniform loop
        const int ibase = start + c * 16;
        // This lane's A-row item (clamped for the ragged tail chunk; results
        // for clamped rows are computed but never written).
        int idx  = ibase + row;
        int item = sorted_idx[idx < end ? idx : start];
        const float* hp = entity_emb + (size_t)head[item] * E_DIM;
        const float* tp = entity_emb + (size_t)tail[item] * E_DIM;

        v8f acc0 = {}, acc1 = {}, acc2 = {}, acc3 = {};

        #pragma unroll 4
        for (int k0 = 0; k0 < E_DIM; k0 += 4) {
            const int k = k0 + khalf * 2;
            // A fragment: a.x=A[row][k], a.y=A[row][k+1], A = h - t
            float2 hv = *(const float2*)(hp + k);
            float2 tv = *(const float2*)(tp + k);
            v2f a; a.x = hv.x - tv.x; a.y = hv.y - tv.y;
            // B fragments from LDS: b.x=M[k][col], b.y=M[k+1][col], col = nt*16+row
            const float* m0 = Ms + k * R_DIM + row;
            const float* m1 = m0 + R_DIM;
            v2f b0, b1, b2, b3;
            b0.x = m0[0];  b0.y = m1[0];
            b1.x = m0[16]; b1.y = m1[16];
            b2.x = m0[32]; b2.y = m1[32];
            b3.x = m0[48]; b3.y = m1[48];
            acc0 = __builtin_amdgcn_wmma_f32_16x16x4_f32(false, a, false, b0, (short)0, acc0, false, false);
            acc1 = __builtin_amdgcn_wmma_f32_16x16x4_f32(false, a, false, b1, (short)0, acc1, false, false);
            acc2 = __builtin_amdgcn_wmma_f32_16x16x4_f32(false, a, false, b2, (short)0, acc2, false, false);
            acc3 = __builtin_amdgcn_wmma_f32_16x16x4_f32(false, a, false, b3, (short)0, acc3, false, false);
        }

        // C layout: VGPR j holds row (j + khalf*8), this lane's column is
        // nt*16 + row. Add r, abs, then reduce the 16 columns across the
        // 16-lane half-wave (xor masks < 16 stay within the half).
        float partial[8];
        #pragma unroll
        for (int j = 0; j < 8; ++j) {
            float s = fabsf(acc0[j] + Rs[row +  0])
                    + fabsf(acc1[j] + Rs[row + 16])
                    + fabsf(acc2[j] + Rs[row + 32])
                    + fabsf(acc3[j] + Rs[row + 48]);
            #pragma unroll
            for (int m = 1; m < 16; m <<= 1)
                s += __shfl_xor(s, m, 32);
            partial[j] = s;
        }

        // lane 0 holds rows 0..7, lane 16 holds rows 8..15
        if (row == 0) {
            const int rbase = khalf * 8;
            #pragma unroll
            for (int j = 0; j < 8; ++j) {
                int ridx = ibase + rbase + j;
                if (ridx < end) out[sorted_idx[ridx]] = partial[j];
            }
        }
    }
}

// ---------------- host launcher ----------------

extern "C" void kernel_launch(void* const* d_in, const int* in_sizes, int n_in,
                              void* d_out, int out_size, void* d_ws, size_t ws_size,
                              hipStream_t stream) {
    const int*   head         = (const int*)d_in[0];
    const int*   relation     = (const int*)d_in[1];
    const int*   tail         = (const int*)d_in[2];
    const float* entity_emb   = (const float*)d_in[3];
    const float* relation_emb = (const float*)d_in[4];
    const float* proj_emb     = (const float*)d_in[5];
    float*       out          = (float*)d_out;

    const int B     = in_sizes[0];                      // 16384
    const int n_rel = in_sizes[4] / R_DIM;              // 1000

    // workspace layout (ints)
    int* counts  = (int*)d_ws;
    int* cursors = counts + ((n_rel + 255) & ~255);
    int* offsets = cursors + ((n_rel + 255) & ~255);
    int* sorted  = offsets + ((n_rel + 1 + 255) & ~255);

    k_zero   <<<(n_rel + 255) / 256, 256, 0, stream>>>(counts, cursors, n_rel);
    k_hist   <<<(B + 255) / 256,     256, 0, stream>>>(relation, counts, B);
    k_scan   <<<1, 32, 0, stream>>>(counts, offsets, n_rel);
    k_scatter<<<(B + 255) / 256,     256, 0, stream>>>(relation, offsets, cursors, sorted, B);

    transr_wmma<<<n_rel, 256, 0, stream>>>(head, tail, entity_emb, relation_emb,
                                           proj_emb, offsets, sorted, out);
}